// ANFIS_16544214024435
// MI455X (gfx1250) — compile-verified
//
#include <hip/hip_runtime.h>

// ANFIS forward restructured as two WMMA GEMMs on gfx1250 (wave32):
//   S_log = Q(B x 18) @ Sel(18 x 729)   -> strengths = exp2(S_log)   (q pre-scaled by log2e)
//   RO    = [x,1](B x 7) @ W^T(7 x 729)
//   out[b] = sigmoid( sum_r strength*RO / sum_r strength )
// Each wave owns a 16-row batch tile, sweeps 46 column tiles of 16 rules,
// using chained V_WMMA_F32_16X16X4_F32 (5 K-steps for Sel GEMM, 2 for W GEMM).

typedef float v2f __attribute__((ext_vector_type(2)));
typedef float v8f __attribute__((ext_vector_type(8)));

#define ANFIS_B        32768
#define ANFIS_RULES    729      // 3^6
#define ANFIS_COLTILES 46       // ceil(729/16) -> padded to 736
#define ANFIS_LOG2E    1.4426950408889634f

// ---------------------------------------------------------------------------
// Setup: build per-lane WMMA B-operand images shared by every wave.
// Layout: float2 bops[(tile*7 + j)*32 + lane]
//   j=0..4 : Sel (18x736, K padded to 20)  B-operand pairs for K-step j
//   j=5..6 : W^T (8x736, K padded to 8)    B-operand pairs for K-step j-5
// Per ISA 7.12.2 (f32 B 4x16): VGPR0 = {K0 lanes0-15, K2 lanes16-31},
//                              VGPR1 = {K1 lanes0-15, K3 lanes16-31}.
// ---------------------------------------------------------------------------
__global__ void anfis_setup_bops(const float* __restrict__ rw,
                                 float2* __restrict__ bops) {
  const int t    = blockIdx.x;        // column tile
  const int lane = threadIdx.x;       // 0..31 (one wave per block)
  const int n    = lane & 15;
  const int hi   = lane >> 4;
  const int col  = t * 16 + n;        // rule index (may be >= 729 pad)
  const int p3[6] = {1, 3, 9, 27, 81, 243};

  // Selection-matrix steps: Sel[k][r] = (rules[r][i] == m), k = i*3+m,
  // rules[r][i] = (r / 3^(5-i)) % 3  (meshgrid 'ij' lexicographic order).
#pragma unroll
  for (int j = 0; j < 5; ++j) {
    const int k0 = 4 * j + 2 * hi;
    float v[2];
#pragma unroll
    for (int c = 0; c < 2; ++c) {
      const int k = k0 + c;
      float sv = 0.0f;
      if (k < 18 && col < ANFIS_RULES) {
        const int i   = k / 3;
        const int m   = k % 3;
        const int dig = (col / p3[5 - i]) % 3;
        sv = (dig == m) ? 1.0f : 0.0f;
      }
      v[c] = sv;
    }
    float2 f2; f2.x = v[0]; f2.y = v[1];
    bops[(t * 7 + j) * 32 + lane] = f2;
  }

  // Weight steps: W^T[k][r] = rule_weights[r*7 + k], k<7; zero pad k=7, r>=729.
#pragma unroll
  for (int j = 5; j < 7; ++j) {
    const int k0 = 4 * (j - 5) + 2 * hi;
    float v[2];
#pragma unroll
    for (int c = 0; c < 2; ++c) {
      const int k = k0 + c;
      float wv = 0.0f;
      if (k < 7 && col < ANFIS_RULES) wv = rw[col * 7 + k];
      v[c] = wv;
    }
    float2 f2; f2.x = v[0]; f2.y = v[1];
    bops[(t * 7 + j) * 32 + lane] = f2;
  }
}

// ---------------------------------------------------------------------------
// Main kernel: one wave per 16-row batch tile (2048 waves total).
// ---------------------------------------------------------------------------
__global__ void __launch_bounds__(256)
anfis_wmma_main(const float* __restrict__ x,
                const float* __restrict__ centers,
                const float* __restrict__ sigmas,
                const float2* __restrict__ bops,
                float* __restrict__ out) {
  const int lane     = threadIdx.x & 31;
  const int wave     = (blockIdx.x * blockDim.x + threadIdx.x) >> 5;
  const int row_base = wave * 16;
  const int hi       = lane >> 4;
  const int row      = row_base + (lane & 15);

  // Per-row inputs (A-matrix row M = lane%16).
  float xv[6];
#pragma unroll
  for (int i = 0; i < 6; ++i) xv[i] = x[row * 6 + i];

  // q[k] = -0.5*log2e * ((x_i - c_{i,m})/s_{i,m})^2, k = i*3+m; pad k=18,19.
  float q[20];
#pragma unroll
  for (int i = 0; i < 6; ++i) {
#pragma unroll
    for (int m = 0; m < 3; ++m) {
      const float c = centers[i * 3 + m];
      const float s = sigmas[i * 3 + m];
      const float z = (xv[i] - c) / s;
      q[i * 3 + m] = z * z * (-0.5f * ANFIS_LOG2E);
    }
  }
  q[18] = 0.0f; q[19] = 0.0f;

  // A-operands (ISA 7.12.2 f32 A 16x4): lane<16 -> (K0,K1), lane>=16 -> (K2,K3).
  v2f aQ[5];
#pragma unroll
  for (int s = 0; s < 5; ++s) {
    aQ[s].x = hi ? q[4 * s + 2] : q[4 * s + 0];
    aQ[s].y = hi ? q[4 * s + 3] : q[4 * s + 1];
  }
  v2f aW[2];
  aW[0].x = hi ? xv[2] : xv[0];
  aW[0].y = hi ? xv[3] : xv[1];
  aW[1].x = hi ? 1.0f  : xv[4];   // k=6 is the bias column of [x, 1]
  aW[1].y = hi ? 0.0f  : xv[5];   // k=7 zero pad

  float sumS[8] = {0, 0, 0, 0, 0, 0, 0, 0};
  float sumN[8] = {0, 0, 0, 0, 0, 0, 0, 0};

  const float cmask = ((lane & 15) < (ANFIS_RULES - (ANFIS_COLTILES - 1) * 16))
                          ? 1.0f : 0.0f;  // valid cols in the last tile (729%16=9)

#pragma unroll 1
  for (int t = 0; t < ANFIS_COLTILES; ++t) {
    const float2* bp = bops + (size_t)(t * 7) * 32 + lane;
    v2f b[7];
#pragma unroll
    for (int j = 0; j < 7; ++j) {
      const float2 f2 = bp[j * 32];
      b[j].x = f2.x; b[j].y = f2.y;
    }

    // Sel GEMM: 5 chained f32 WMMAs (K = 20)
    v8f accS = {0, 0, 0, 0, 0, 0, 0, 0};
#pragma unroll
    for (int s = 0; s < 5; ++s)
      accS = __builtin_amdgcn_wmma_f32_16x16x4_f32(
          false, aQ[s], false, b[s], (short)0, accS, false, false);

    // Weight GEMM: 2 chained f32 WMMAs (K = 8)
    v8f accW = {0, 0, 0, 0, 0, 0, 0, 0};
#pragma unroll
    for (int s = 0; s < 2; ++s)
      accW = __builtin_amdgcn_wmma_f32_16x16x4_f32(
          false, aW[s], false, b[5 + s], (short)0, accW, false, false);

    const float cm = (t == ANFIS_COLTILES - 1) ? cmask : 1.0f;
#pragma unroll
    for (int e = 0; e < 8; ++e) {
      const float st = __builtin_amdgcn_exp2f(accS[e]) * cm;  // v_exp_f32
      sumS[e] += st;
      sumN[e] += st * accW[e];
    }
  }

  // Reduce over the 16 lanes sharing each C/D row, then sigmoid + store.
  // C/D element e of lane L holds row (e + 8*(L>=16)), col (L&15).
#pragma unroll
  for (int e = 0; e < 8; ++e) {
    float den = sumS[e];
    float num = sumN[e];
#pragma unroll
    for (int m = 8; m >= 1; m >>= 1) {
      den += __shfl_xor(den, m, 16);
      num += __shfl_xor(num, m, 16);
    }
    const float r   = num / den;
    const float sig = 1.0f / (1.0f + __builtin_amdgcn_exp2f(-r * ANFIS_LOG2E));
    if ((lane & 15) == e) out[row_base + e + 8 * hi] = sig;
  }
}

// ---------------------------------------------------------------------------
extern "C" void kernel_launch(void* const* d_in, const int* in_sizes, int n_in,
                              void* d_out, int out_size, void* d_ws, size_t ws_size,
                              hipStream_t stream) {
  (void)in_sizes; (void)n_in; (void)out_size; (void)ws_size;
  const float* x       = (const float*)d_in[0];  // (32768, 6)
  const float* centers = (const float*)d_in[1];  // (6, 3)
  const float* sigmas  = (const float*)d_in[2];  // (6, 3)
  const float* rw      = (const float*)d_in[3];  // (729, 7)
  float* out           = (float*)d_out;          // (32768, 1)
  float2* bops         = (float2*)d_ws;          // needs 46*7*32*8 = 82,432 B

  anfis_setup_bops<<<ANFIS_COLTILES, 32, 0, stream>>>(rw, bops);

  // 32768 rows / 16 per wave = 2048 waves; 8 waves (256 thr) per block.
  anfis_wmma_main<<<ANFIS_B / 16 / 8, 256, 0, stream>>>(x, centers, sigmas, bops, out);
}